// AudioModelV3_30709016166496
// MI455X (gfx1250) — compile-verified
//
#include <hip/hip_runtime.h>
#include <hip/hip_bf16.h>

// ---------- types / wmma ----------
typedef float f32x2 __attribute__((ext_vector_type(2)));
typedef float f32x8 __attribute__((ext_vector_type(8)));

static __device__ inline f32x8 wmma4(f32x2 a, f32x2 b, f32x8 c) {
  // D(16x16,f32) = A(16x4,f32) x B(4x16,f32) + C
  return __builtin_amdgcn_wmma_f32_16x16x4_f32(false, a, false, b, (short)0, c,
                                               false, false);
}

// ---------- math helpers ----------
static __device__ inline float dev_logsig(float x) {
  return fminf(x, 0.f) - log1pf(__expf(-fabsf(x)));
}
static __device__ inline float dev_sigmoid(float x) {
  return 1.f / (1.f + __expf(-x));
}
static __device__ inline float dev_silu(float x) { return x * dev_sigmoid(x); }
static __device__ inline float dev_gelu_tanh(float x) {
  float x3 = x * x * x;
  return 0.5f * x * (1.f + tanhf(0.7978845608028654f * (x + 0.044715f * x3)));
}

// =====================================================================
// f32 WMMA GEMM, register-blocked: one wave computes a 16x64 output tile
// (4 accumulators share each A fragment -> 4 wmma per A load pair).
// C[M,Nc] = A[M,K] @ B[K,Nc] (+bias) (+=C if acc).
// Requires M%16==0, K%4==0, Nc%64==0 (true for all call sites).
// =====================================================================
__global__ __launch_bounds__(256) void k_gemm(const float* __restrict__ A,
                                              const float* __restrict__ Bm,
                                              const float* __restrict__ bias,
                                              float* __restrict__ C, int M,
                                              int K, int Nc, int acc) {
  int wave = blockIdx.x * (blockDim.x >> 5) + (threadIdx.x >> 5);
  int tilesN = Nc >> 6;  // 64-wide N blocks
  int tilesM = M >> 4;
  if (wave >= tilesM * tilesN) return;
  int lane = threadIdx.x & 31;
  int lr = lane & 15, hi = lane >> 4;
  int tm = wave / tilesN, tn = wave % tilesN;
  int m0 = tm << 4, n0 = tn << 6;
  f32x8 c0 = {0.f, 0.f, 0.f, 0.f, 0.f, 0.f, 0.f, 0.f};
  f32x8 c1 = c0, c2 = c0, c3 = c0;
  const float* arow = A + (size_t)(m0 + lr) * K;
  const float* b0 = Bm + n0 + lr;
  const float* b1 = b0 + 16;
  const float* b2 = b0 + 32;
  const float* b3 = b0 + 48;
  for (int kk = 0; kk < K; kk += 4) {
    int ka = kk + (hi << 1);
    __builtin_prefetch(arow + ka + 64, 0, 1);  // global_prefetch_b8
    f32x2 a;
    a.x = arow[ka];
    a.y = arow[ka + 1];
    size_t r0 = (size_t)ka * Nc, r1 = (size_t)(ka + 1) * Nc;
    f32x2 bb;
    bb.x = b0[r0]; bb.y = b0[r1];
    c0 = wmma4(a, bb, c0);
    bb.x = b1[r0]; bb.y = b1[r1];
    c1 = wmma4(a, bb, c1);
    bb.x = b2[r0]; bb.y = b2[r1];
    c2 = wmma4(a, bb, c2);
    bb.x = b3[r0]; bb.y = b3[r1];
    c3 = wmma4(a, bb, c3);
  }
  for (int j = 0; j < 4; ++j) {
    f32x8 cj = (j == 0) ? c0 : (j == 1) ? c1 : (j == 2) ? c2 : c3;
    int col = n0 + (j << 4) + lr;
    float bv = bias ? bias[col] : 0.f;
    for (int i = 0; i < 8; ++i) {
      int r = m0 + (hi << 3) + i;
      size_t o = (size_t)r * Nc + col;
      float v = cj[i] + bv;
      if (acc) v += C[o];
      C[o] = v;
    }
  }
}

// =====================================================================
// LayerNorm over 256, one wave per token. mode 1 => apply SELU after.
// =====================================================================
__global__ __launch_bounds__(256) void k_ln256(const float* __restrict__ x,
                                               const float* __restrict__ w,
                                               float* __restrict__ out,
                                               int Ntok, int mode) {
  int wave = blockIdx.x * (blockDim.x >> 5) + (threadIdx.x >> 5);
  if (wave >= Ntok) return;
  int lane = threadIdx.x & 31;
  const float* xp = x + (size_t)wave * 256;
  float v[8], s = 0.f, s2 = 0.f;
  for (int i = 0; i < 8; ++i) {
    v[i] = xp[lane + 32 * i];
    s += v[i];
    s2 += v[i] * v[i];
  }
  for (int m = 16; m >= 1; m >>= 1) {
    s += __shfl_xor(s, m);
    s2 += __shfl_xor(s2, m);
  }
  float mu = s * (1.f / 256.f);
  float var = s2 * (1.f / 256.f) - mu * mu;
  float rs = rsqrtf(var + 1e-5f);
  for (int i = 0; i < 8; ++i) {
    int cix = lane + 32 * i;
    float y = (v[i] - mu) * rs * w[cix];
    if (mode == 1) {
      y = 1.0507009873554805f *
          (y > 0.f ? y : 1.6732632423543772f * (__expf(y) - 1.f));
    }
    out[(size_t)wave * 256 + cix] = y;
  }
}

// =====================================================================
// Multi-head layernorm: one wave per (token, head). dh in {64,128}.
// addout=1 accumulates into out (residual), else overwrites.
// =====================================================================
__global__ __launch_bounds__(256) void k_mhln(const float* __restrict__ x,
                                              const float* __restrict__ w,
                                              float* __restrict__ out, int Ntok,
                                              int nh, int dh, int addout) {
  int wave = blockIdx.x * (blockDim.x >> 5) + (threadIdx.x >> 5);
  if (wave >= Ntok * nh) return;
  int lane = threadIdx.x & 31;
  int tok = wave / nh, h = wave % nh;
  int width = nh * dh;
  const float* xp = x + (size_t)tok * width + h * dh;
  int np = dh >> 5;
  float v[4], s = 0.f, s2 = 0.f;
  for (int i = 0; i < np; ++i) {
    v[i] = xp[lane + 32 * i];
    s += v[i];
    s2 += v[i] * v[i];
  }
  for (int m = 16; m >= 1; m >>= 1) {
    s += __shfl_xor(s, m);
    s2 += __shfl_xor(s2, m);
  }
  float inv = 1.f / (float)dh;
  float mu = s * inv;
  float var = s2 * inv - mu * mu;
  float rs = rsqrtf(var + 1e-5f);
  for (int i = 0; i < np; ++i) {
    int cix = lane + 32 * i;
    float y = (v[i] - mu) * rs * w[h * dh + cix];
    size_t o = (size_t)tok * width + h * dh + cix;
    if (addout)
      out[o] += y;
    else
      out[o] = y;
  }
}

// =====================================================================
// Depthwise causal conv (K=4) + SiLU
// =====================================================================
__global__ void k_conv(const float* __restrict__ in, const float* __restrict__ w,
                       const float* __restrict__ bias, float* __restrict__ out,
                       int Bn, int S, int C, int ldin, int ldout) {
  int idx = blockIdx.x * blockDim.x + threadIdx.x;
  if (idx >= Bn * S * C) return;
  int tok = idx / C, c = idx % C;
  int b = tok / S, s = tok % S;
  float acc = bias[c];
  for (int i = 0; i < 4; ++i) {
    int t = s - 3 + i;
    if (t >= 0) acc += in[(size_t)(b * S + t) * ldin + c] * w[c * 4 + i];
  }
  out[(size_t)tok * ldout + c] = dev_silu(acc);
}

// =====================================================================
// headwise mix: 128 heads of 4x4 (q/k/v projections)
// =====================================================================
__global__ void k_hw(const float* __restrict__ in, int ldin,
                     const float* __restrict__ w, float* __restrict__ out,
                     int Ntok) {
  int idx = blockIdx.x * blockDim.x + threadIdx.x;
  if (idx >= Ntok * 512) return;
  int tok = idx >> 9, j = idx & 511;
  int hh = j >> 2, o = j & 3;
  const float* ip = in + (size_t)tok * ldin + (hh << 2);
  const float* wp = w + ((hh << 2) + o) * 4;
  float acc = ip[0] * wp[0] + ip[1] * wp[1] + ip[2] * wp[2] + ip[3] * wp[3];
  out[(size_t)tok * 512 + j] = acc;
}

// =====================================================================
// mLSTM input/forget gate GEMVs over concat(q,k,v) [1536] -> per head
// one wave per (token, head); output layout [b, h, s]
// =====================================================================
__global__ __launch_bounds__(256) void k_gates(
    const float* __restrict__ Q, const float* __restrict__ Kb,
    const float* __restrict__ V, const float* __restrict__ igw,
    const float* __restrict__ igb, const float* __restrict__ fgw,
    const float* __restrict__ fgb, float* __restrict__ igo,
    float* __restrict__ fgo, int Ntok, int S) {
  int wave = blockIdx.x * (blockDim.x >> 5) + (threadIdx.x >> 5);
  if (wave >= Ntok * 4) return;
  int lane = threadIdx.x & 31;
  int tok = wave >> 2, h = wave & 3;
  const float* qp = Q + (size_t)tok * 512;
  const float* kp = Kb + (size_t)tok * 512;
  const float* vp = V + (size_t)tok * 512;
  const float* iw = igw + h * 1536;
  const float* fw = fgw + h * 1536;
  float ai = 0.f, af = 0.f;
  for (int j = lane; j < 512; j += 32) {
    float q = qp[j], k = kp[j], v = vp[j];
    ai += q * iw[j] + k * iw[512 + j] + v * iw[1024 + j];
    af += q * fw[j] + k * fw[512 + j] + v * fw[1024 + j];
  }
  for (int m = 16; m >= 1; m >>= 1) {
    ai += __shfl_xor(ai, m);
    af += __shfl_xor(af, m);
  }
  if (lane == 0) {
    int b = tok / S, s = tok % S;
    igo[(size_t)(b * 4 + h) * S + s] = ai + igb[h];
    fgo[(size_t)(b * 4 + h) * S + s] = af + fgb[h];
  }
}

// =====================================================================
// per (b,h): lfc cumsum of log-sigmoid(fg), and maxD[r] via prefix max
// =====================================================================
__global__ void k_prefix(const float* __restrict__ igo,
                         const float* __restrict__ fgo, float* __restrict__ lfc,
                         float* __restrict__ maxD, int BH, int S) {
  int id = blockIdx.x * blockDim.x + threadIdx.x;
  if (id >= BH) return;
  const float* fp = fgo + (size_t)id * S;
  const float* ip = igo + (size_t)id * S;
  float* lp = lfc + (size_t)id * (S + 1);
  float* mp = maxD + (size_t)id * S;
  float run = 0.f, pm = -1e30f;
  lp[0] = 0.f;
  for (int t = 0; t < S; ++t) {
    run += dev_logsig(fp[t]);
    lp[t + 1] = run;
    float g = ip[t] - run;
    pm = fmaxf(pm, g);
    mp[t] = run + pm;  // maxD[r] = lfc[r+1] + max_{t<=r}(ig[t]-lfc[t+1])
  }
}

// =====================================================================
// mLSTM decay attention, one workgroup per (b,head).
// Full 208x208 score matrix in LDS. Both GEMMs on f32 WMMA.
// Out aliases Q (written only after phase-1 reads complete).
// =====================================================================
__global__ __launch_bounds__(256) void k_attn(
    const float* __restrict__ Q, const float* __restrict__ Kc,
    const float* __restrict__ V, const float* __restrict__ ig,
    const float* __restrict__ lfc, const float* __restrict__ maxD,
    float* __restrict__ Out, int S) {
  __shared__ float sc[208 * 208];
  __shared__ float inv_norm[208];
  int bh = blockIdx.x;
  int b = bh >> 2, h = bh & 3;
  const float* igp = ig + (size_t)bh * S;
  const float* lfcp = lfc + (size_t)bh * (S + 1);
  const float* mdp = maxD + (size_t)bh * S;
  const size_t baseQ = ((size_t)b * S) * 512 + h * 128;
  int tid = threadIdx.x;
  int wave = tid >> 5, lane = tid & 31, lr = lane & 15, hi = lane >> 4;
  for (int i = tid; i < 208 * 208; i += 256) sc[i] = 0.f;
  __syncthreads();
  const float inv_sqrt_d = 0.08838834764831845f;  // 1/sqrt(128)

  // Phase 1: weighted scores for the 91 lower-triangular 16x16 tiles
  for (int t = wave; t < 91; t += 8) {
    int tr = 0;
    while (((tr + 1) * (tr + 2)) / 2 <= t) ++tr;
    int tc = t - (tr * (tr + 1)) / 2;
    int m0 = tr << 4, n0 = tc << 4;
    f32x8 c = {0.f, 0.f, 0.f, 0.f, 0.f, 0.f, 0.f, 0.f};
    int rA = m0 + lr;
    int tB = n0 + lr;
    const float* qrow = Q + baseQ + (size_t)rA * 512;
    const float* krow = Kc + baseQ + (size_t)tB * 512;
    bool va = rA < S, vb = tB < S;
    for (int kk = 0; kk < 128; kk += 4) {
      int ka = kk + (hi << 1);
      f32x2 a, bb;
      a.x = va ? qrow[ka] : 0.f;
      a.y = va ? qrow[ka + 1] : 0.f;
      bb.x = vb ? krow[ka] : 0.f;
      bb.y = vb ? krow[ka + 1] : 0.f;
      c = wmma4(a, bb, c);
    }
    int tcol = n0 + lr;
    for (int i = 0; i < 8; ++i) {
      int r = m0 + (hi << 3) + i;
      if (r < S) {
        float v = 0.f;
        if (tcol <= r && tcol < S) {
          float d = lfcp[r + 1] - lfcp[tcol + 1] + igp[tcol] - mdp[r];
          v = c[i] * inv_sqrt_d * __expf(d);
        }
        sc[r * 208 + tcol] = v;
      }
    }
  }
  __syncthreads();

  // Phase 2: per-row normalization
  for (int r = tid; r < 208; r += 256) {
    if (r < S) {
      float s = 0.f;
      for (int t2 = 0; t2 <= r; ++t2) s += sc[r * 208 + t2];
      float nrm = fmaxf(fabsf(s), __expf(-mdp[r])) + 1e-6f;
      inv_norm[r] = 1.f / nrm;
    } else
      inv_norm[r] = 0.f;
  }
  __syncthreads();

  // Phase 3: out = (scores * inv_norm) @ V, WMMA over K = t
  for (int t = wave; t < 13 * 8; t += 8) {
    int tr = t >> 3, nc = t & 7;
    int m0 = tr << 4, n0 = nc << 4;
    f32x8 c = {0.f, 0.f, 0.f, 0.f, 0.f, 0.f, 0.f, 0.f};
    int rA = m0 + lr;
    float sA = inv_norm[rA];
    int kmax = m0 + 16;  // strictly causal: t <= r < m0+16
    int dcol = n0 + lr;
    for (int kk = 0; kk < kmax; kk += 4) {
      int ka = kk + (hi << 1);
      f32x2 a, bb;
      a.x = sc[rA * 208 + ka] * sA;
      a.y = sc[rA * 208 + ka + 1] * sA;
      bb.x = (ka < S) ? V[baseQ + (size_t)ka * 512 + dcol] : 0.f;
      bb.y = (ka + 1 < S) ? V[baseQ + (size_t)(ka + 1) * 512 + dcol] : 0.f;
      c = wmma4(a, bb, c);
    }
    for (int i = 0; i < 8; ++i) {
      int r = m0 + (hi << 3) + i;
      if (r < S) Out[baseQ + (size_t)r * 512 + dcol] = c[i];
    }
  }
}

// =====================================================================
// h_state = (hn + skip*xca) * silu(z)
// =====================================================================
__global__ void k_hstate(const float* __restrict__ hn,
                         const float* __restrict__ xca,
                         const float* __restrict__ xi,  // z = xi[:,512:1024]
                         const float* __restrict__ skip,
                         float* __restrict__ out, int Ntok) {
  int idx = blockIdx.x * blockDim.x + threadIdx.x;
  if (idx >= Ntok * 512) return;
  int tok = idx >> 9, c = idx & 511;
  float z = xi[(size_t)tok * 1024 + 512 + c];
  out[idx] = (hn[idx] + skip[c] * xca[idx]) * dev_silu(z);
}

// =====================================================================
// sLSTM Wx: Wx[tok, g*256 + hd*64 + o]; g in {i,f} from xc, {z,o} from xn
// =====================================================================
__global__ void k_swx(const float* __restrict__ xc, const float* __restrict__ xn,
                      const float* __restrict__ gw, float* __restrict__ out,
                      int Ntok) {
  int idx = blockIdx.x * blockDim.x + threadIdx.x;
  if (idx >= Ntok * 1024) return;
  int tok = idx >> 10, r = idx & 1023;
  int g = r >> 8, hd = (r >> 6) & 3, o = r & 63;
  const float* src = (g < 2) ? xc : xn;
  const float* sp = src + (size_t)tok * 256 + (hd << 6);
  const float* wp = gw + ((((g << 2) + hd) << 6) + o) * 64;
  float acc = 0.f;
  for (int d = 0; d < 64; ++d) acc += sp[d] * wp[d];
  out[idx] = acc;
}

// =====================================================================
// sLSTM recurrent scan: one workgroup per (b, head), 64 threads.
// Recurrent weight slice (4x64x64 = 64KB) cached in LDS, h broadcast in LDS.
// =====================================================================
__global__ __launch_bounds__(64) void k_sscan(const float* __restrict__ Wx,
                                              const float* __restrict__ recw,
                                              const float* __restrict__ bias,
                                              float* __restrict__ ys, int Bn,
                                              int S) {
  __shared__ float rec[4 * 64 * 64];
  __shared__ float hb[64];
  int blk = blockIdx.x;
  int b = blk >> 2, hd = blk & 3;
  int o = threadIdx.x;
  for (int idx = o; idx < 4 * 64 * 64; idx += 64) {
    int g = idx >> 12, rem = idx & 4095;
    int oo = rem >> 6, d = rem & 63;
    rec[idx] = recw[(((size_t)((g << 2) + hd) << 6) + oo) * 64 + d];
  }
  float bg[4];
  for (int g = 0; g < 4; ++g) bg[g] = bias[(((g << 2) + hd) << 6) + o];
  hb[o] = 0.f;
  float c = 0.f, n = 0.f, m = 0.f;
  __syncthreads();
  for (int s = 0; s < S; ++s) {
    size_t base = ((size_t)(b * S + s)) * 1024 + (hd << 6) + o;
    float raw[4];
    for (int g = 0; g < 4; ++g) {
      float acc = Wx[base + (g << 8)] + bg[g];
      const float* rp = rec + (g << 12) + (o << 6);
      for (int d = 0; d < 64; ++d) acc += hb[d] * rp[d];
      raw[g] = acc;
    }
    float lfm = m + dev_logsig(raw[1]);
    float mn = fmaxf(raw[0], lfm);
    float igt = __expf(raw[0] - mn);
    float fgt = __expf(lfm - mn);
    c = fgt * c + igt * tanhf(raw[2]);
    n = fgt * n + igt;
    float hnew = dev_sigmoid(raw[3]) * (c / n);
    ys[((size_t)(b * S + s)) * 256 + (hd << 6) + o] = hnew;
    m = mn;
    __syncthreads();
    hb[o] = hnew;
    __syncthreads();
  }
}

// =====================================================================
// gated FFN mid: out[tok, j<384] = gelu(up[:, :384]) * up[:, 384:]
// =====================================================================
__global__ void k_mid(const float* __restrict__ up, float* __restrict__ out,
                      int Ntok) {
  int idx = blockIdx.x * blockDim.x + threadIdx.x;
  if (idx >= Ntok * 384) return;
  int tok = idx / 384, j = idx % 384;
  float g = up[(size_t)tok * 768 + j];
  float u = up[(size_t)tok * 768 + 384 + j];
  out[(size_t)tok * 384 + j] = dev_gelu_tanh(g) * u;
}

// ---------- mean pool + heads ----------
__global__ void k_pool(const float* __restrict__ L, float* __restrict__ pooled,
                       int Bn, int S) {
  int idx = blockIdx.x * blockDim.x + threadIdx.x;
  if (idx >= Bn * 256) return;
  int b = idx >> 8, e = idx & 255;
  float s = 0.f;
  for (int t = 0; t < S; ++t) s += L[(size_t)(b * S + t) * 256 + e];
  pooled[idx] = s / (float)S;
}

__global__ void k_heads(const float* __restrict__ pooled,
                        const float* __restrict__ we, const float* __restrict__ be,
                        const float* __restrict__ wsn,
                        const float* __restrict__ bsn, float* __restrict__ out,
                        int Bn) {
  int idx = blockIdx.x * blockDim.x + threadIdx.x;
  if (idx >= Bn * 10) return;
  int b = idx / 10, j = idx % 10;
  const float* pp = pooled + (size_t)b * 256;
  if (j < 7) {
    float acc = be[j];
    for (int e = 0; e < 256; ++e) acc += pp[e] * we[e * 7 + j];
    out[b * 7 + j] = acc;
  } else {
    int jj = j - 7;
    float acc = bsn[jj];
    for (int e = 0; e < 256; ++e) acc += pp[e] * wsn[e * 3 + jj];
    out[64 * 7 + b * 3 + jj] = acc;
  }
}

// =====================================================================
// host orchestration
// =====================================================================
extern "C" void kernel_launch(void* const* d_in, const int* in_sizes, int n_in,
                              void* d_out, int out_size, void* d_ws,
                              size_t ws_size, hipStream_t stream) {
  (void)in_sizes; (void)n_in; (void)out_size; (void)ws_size;
  const int Bn = 64, S = 199, N = Bn * S;

  const float* x = (const float*)d_in[0];
  const float* w_down = (const float*)d_in[1];
  const float* b_down = (const float*)d_in[2];
  const float* m_ln_w = (const float*)d_in[3];
  const float* m_proj_up = (const float*)d_in[4];
  const float* m_conv_w = (const float*)d_in[5];
  const float* m_conv_b = (const float*)d_in[6];
  const float* m_q_w = (const float*)d_in[7];
  const float* m_k_w = (const float*)d_in[8];
  const float* m_v_w = (const float*)d_in[9];
  const float* m_ig_w = (const float*)d_in[10];
  const float* m_ig_b = (const float*)d_in[11];
  const float* m_fg_w = (const float*)d_in[12];
  const float* m_fg_b = (const float*)d_in[13];
  const float* m_skip = (const float*)d_in[14];
  const float* m_outnorm_w = (const float*)d_in[15];
  const float* m_proj_down = (const float*)d_in[16];
  const float* s_ln_w = (const float*)d_in[17];
  const float* s_conv_w = (const float*)d_in[18];
  const float* s_conv_b = (const float*)d_in[19];
  const float* s_gate_w = (const float*)d_in[20];
  const float* s_rec_w = (const float*)d_in[21];
  const float* s_bias = (const float*)d_in[22];
  const float* s_gn_w = (const float*)d_in[23];
  const float* s_ln2_w = (const float*)d_in[24];
  const float* s_ffn_up = (const float*)d_in[25];
  const float* s_ffn_down = (const float*)d_in[26];
  const float* post_ln_w = (const float*)d_in[27];
  const float* w_emo = (const float*)d_in[28];
  const float* b_emo = (const float*)d_in[29];
  const float* w_sen = (const float*)d_in[30];
  const float* b_sen = (const float*)d_in[31];

  float* ws = (float*)d_ws;
  float* h = ws;
  float* L = h + (size_t)N * 256;
  float* Abuf = L + (size_t)N * 256;
  float* X = Abuf + (size_t)N * 1024;
  float* Q = X + (size_t)N * 512;
  float* Kb = Q + (size_t)N * 512;
  float* V = Kb + (size_t)N * 512;
  float* ig = V + (size_t)N * 512;
  float* fg = ig + (size_t)Bn * 4 * S;
  float* lfc = fg + (size_t)Bn * 4 * S;
  float* mdb = lfc + (size_t)Bn * 4 * (S + 1);
  float* pooled = mdb + (size_t)Bn * 4 * S;

  auto gemm = [&](const float* A, const float* Bm, const float* bias, float* C,
                  int M, int K, int Nc, int acc) {
    int tiles = (M / 16) * (Nc / 64);
    int blocks = (tiles + 7) / 8;
    k_gemm<<<blocks, 256, 0, stream>>>(A, Bm, bias, C, M, K, Nc, acc);
  };
  auto ewblocks = [](int total) { return (total + 255) / 256; };
  auto wvblocks = [](int waves) { return (waves + 7) / 8; };

  // h = x @ w_down + b_down
  gemm(x, w_down, b_down, h, N, 1024, 256, 0);

  int mi = 0, si = 0;
  for (int blk = 0; blk < 7; ++blk) {
    if (blk == 1 || blk == 4) {
      // ---------------- sLSTM block ----------------
      k_ln256<<<wvblocks(N), 256, 0, stream>>>(h, s_ln_w + si * 256, L, N, 0);
      k_conv<<<ewblocks(N * 256), 256, 0, stream>>>(
          L, s_conv_w + si * 1024, s_conv_b + si * 256, X, Bn, S, 256, 256, 256);
      k_swx<<<ewblocks(N * 1024), 256, 0, stream>>>(
          X, L, s_gate_w + (size_t)si * 65536, Abuf, N);
      k_sscan<<<Bn * 4, 64, 0, stream>>>(Abuf, s_rec_w + (size_t)si * 65536,
                                         s_bias + si * 1024, Q, Bn, S);
      k_mhln<<<wvblocks(N * 4), 256, 0, stream>>>(Q, s_gn_w + si * 256, h, N, 4,
                                                  64, 1);
      // gated FFN
      k_ln256<<<wvblocks(N), 256, 0, stream>>>(h, s_ln2_w + si * 256, L, N, 0);
      gemm(L, s_ffn_up + (size_t)si * 256 * 768, nullptr, Abuf, N, 256, 768, 0);
      k_mid<<<ewblocks(N * 384), 256, 0, stream>>>(Abuf, X, N);
      gemm(X, s_ffn_down + (size_t)si * 384 * 256, nullptr, h, N, 384, 256, 1);
      ++si;
    } else {
      // ---------------- mLSTM block ----------------
      k_ln256<<<wvblocks(N), 256, 0, stream>>>(h, m_ln_w + mi * 256, L, N, 0);
      gemm(L, m_proj_up + (size_t)mi * 256 * 1024, nullptr, Abuf, N, 256, 1024,
           0);
      k_conv<<<ewblocks(N * 512), 256, 0, stream>>>(
          Abuf, m_conv_w + mi * 2048, m_conv_b + mi * 512, X, Bn, S, 512, 1024,
          512);
      k_hw<<<ewblocks(N * 512), 256, 0, stream>>>(X, 512, m_q_w + mi * 2048, Q,
                                                  N);
      k_hw<<<ewblocks(N * 512), 256, 0, stream>>>(X, 512, m_k_w + mi * 2048, Kb,
                                                  N);
      k_hw<<<ewblocks(N * 512), 256, 0, stream>>>(Abuf, 1024, m_v_w + mi * 2048,
                                                  V, N);
      k_gates<<<wvblocks(N * 4), 256, 0, stream>>>(
          Q, Kb, V, m_ig_w + mi * 6144, m_ig_b + mi * 4, m_fg_w + mi * 6144,
          m_fg_b + mi * 4, ig, fg, N, S);
      k_prefix<<<1, 256, 0, stream>>>(ig, fg, lfc, mdb, Bn * 4, S);
      k_attn<<<Bn * 4, 256, 0, stream>>>(Q, Kb, V, ig, lfc, mdb, Q, S);
      k_mhln<<<wvblocks(N * 4), 256, 0, stream>>>(Q, m_outnorm_w + mi * 512, Kb,
                                                  N, 4, 128, 0);
      k_hstate<<<ewblocks(N * 512), 256, 0, stream>>>(Kb, X, Abuf,
                                                      m_skip + mi * 512, V, N);
      gemm(V, m_proj_down + (size_t)mi * 512 * 256, nullptr, h, N, 512, 256, 1);
      ++mi;
    }
  }

  // post LN + SELU, mean pool, heads
  k_ln256<<<wvblocks(N), 256, 0, stream>>>(h, post_ln_w, L, N, 1);
  k_pool<<<ewblocks(Bn * 256), 256, 0, stream>>>(L, pooled, Bn, S);
  k_heads<<<ewblocks(Bn * 10), 256, 0, stream>>>(pooled, w_emo, b_emo, w_sen,
                                                 b_sen, (float*)d_out, Bn);
}